// SerializedEmbedding_8864812498998
// MI455X (gfx1250) — compile-verified
//
#include <hip/hip_runtime.h>
#include <stdint.h>
#include <stddef.h>

// ---------------------------------------------------------------------------
// SerializedEmbedding gather for MI455X (gfx1250).
//
// Reference reduces to: out[t, :] = Wflat[idx[t], :] with Wflat = [49408, 512] f32.
// Pure HBM-bound row gather (~128 MB traffic -> ~6-11 us at 23.3 TB/s).
// Strategy: wave-per-row b128 copies with scalar (SGPR) row addressing,
// RT loads (keep W in 192MB L2), NT streaming stores for the output,
// global_prefetch of the next row to hide random-gather latency.
// ---------------------------------------------------------------------------

#define DIMF      512          // floats per row
#define ROW_F4    (DIMF / 4)   // 128 float4 per row (2048 bytes)
#define THREADS   256          // 8 waves per block (wave32)
#define WAVES_PB  (THREADS / 32)
#define BLOCKS    1024         // 8192 waves -> 4 rows per wave over 32768 rows

typedef float v4f __attribute__((ext_vector_type(4)));

#ifndef __has_builtin
#define __has_builtin(x) 0
#endif

// ---------------------------------------------------------------------------
// Primary kernel: direct global->VGPR->global row copy.
//  - wave index + gathered row index forced uniform via readfirstlane:
//    index fetch becomes s_load_b32, row bases live in SGPRs, and all
//    b128 loads/stores use the saddr+voffset form (voffset = lane*16,
//    512B strides folded into the instruction immediate)
//  - 4x b128 loads per lane -> fully coalesced 512B/instr per wave
//  - NT stores: don't evict W (101 MB, L2-resident) with the streamed output
//  - prefetch next row's 2KB (32 lanes x 64B cachelines) one iteration ahead
// ---------------------------------------------------------------------------
__global__ __launch_bounds__(THREADS) void
gather_rows_direct(const int* __restrict__ idx, const float* __restrict__ W,
                   float* __restrict__ out, int T) {
  const int lane = threadIdx.x & 31;
  // Wave-in-block is uniform across the wave; make it an SGPR value so the
  // whole index/address chain stays on the scalar path.
  const int wvb  = __builtin_amdgcn_readfirstlane((int)(threadIdx.x >> 5));
  const int wave = (int)blockIdx.x * WAVES_PB + wvb;
  const int nwav = (int)gridDim.x * WAVES_PB;

  const v4f* __restrict__ Wv = (const v4f*)W;
  v4f* __restrict__ Ov       = (v4f*)out;

  for (int t = wave; t < T; t += nwav) {
    const int row = __builtin_amdgcn_readfirstlane(idx[t]);  // SGPR row index
    const v4f* __restrict__ src = Wv + (size_t)row * ROW_F4;
    v4f* __restrict__ dst       = Ov + (size_t)t   * ROW_F4;

    // Prefetch the row we'll copy next iteration (covers 2KB via 32x64B).
    const int tn = t + nwav;
    if (tn < T) {
      const int rown = __builtin_amdgcn_readfirstlane(idx[tn]);
      const char* pn = (const char*)(Wv + (size_t)rown * ROW_F4) + (lane << 6);
      __builtin_prefetch(pn, /*rw=*/0, /*locality=*/3);  // -> global_prefetch_b8
    }

    v4f a = src[lane +  0];
    v4f b = src[lane + 32];
    v4f c = src[lane + 64];
    v4f d = src[lane + 96];

    // Streaming (non-temporal) stores: output is write-once, never re-read.
    __builtin_nontemporal_store(a, dst + lane +  0);
    __builtin_nontemporal_store(b, dst + lane + 32);
    __builtin_nontemporal_store(c, dst + lane + 64);
    __builtin_nontemporal_store(d, dst + lane + 96);
  }
}

// ---------------------------------------------------------------------------
// gfx1250 async global<->LDS copy path (ASYNCcnt machinery), kept compiled:
// GLOBAL_LOAD_ASYNC_TO_LDS_B128 / GLOBAL_STORE_ASYNC_FROM_LDS_B128 +
// s_wait_asynccnt. Not launched: per-row it needs two full ASYNCcnt drains
// (mixed load/store counts make pipelined thresholds unsafe, ISA §7.1),
// while the direct path's stores are fire-and-forget.
// ---------------------------------------------------------------------------
#if __has_builtin(__builtin_amdgcn_global_load_async_to_lds_b128) && \
    __has_builtin(__builtin_amdgcn_global_store_async_from_lds_b128)
#define HAVE_ASYNC_LDS 1
#else
#define HAVE_ASYNC_LDS 0
#endif

#if HAVE_ASYNC_LDS

#if __has_builtin(__builtin_amdgcn_s_wait_asynccnt)
#define WAIT_ASYNC(n) __builtin_amdgcn_s_wait_asynccnt(n)
#else
#define WAIT_ASYNC(n) asm volatile("s_wait_asynccnt %0" ::"i"(n) : "memory")
#endif

typedef int v4i __attribute__((vector_size(16)));
typedef __attribute__((address_space(1))) v4i gv4i;  // global 16B vector
typedef __attribute__((address_space(3))) v4i lv4i;  // LDS    16B vector

__global__ __launch_bounds__(THREADS) void
gather_rows_async(const int* __restrict__ idx, const float* __restrict__ W,
                  float* __restrict__ out, int T) {
  __shared__ char smem[WAVES_PB * 2048];  // one 2KB row buffer per wave
  const int lane = threadIdx.x & 31;
  char* buf = smem + (threadIdx.x >> 5) * 2048;

  const int wave = (int)((blockIdx.x * THREADS + threadIdx.x) >> 5);
  const int nwav = (int)((gridDim.x * THREADS) >> 5);

  for (int t = wave; t < T; t += nwav) {
    const char* src = (const char*)W + (size_t)idx[t] * 2048;
    char* dst       = (char*)out     + (size_t)t      * 2048;

#pragma unroll
    for (int k = 0; k < 4; ++k) {
      __builtin_amdgcn_global_load_async_to_lds_b128(
          (gv4i*)(src + ((lane + 32 * k) << 4)),
          (lv4i*)(buf + ((lane + 32 * k) << 4)), 0, 0);
    }
    WAIT_ASYNC(0);  // loads landed in LDS
#pragma unroll
    for (int k = 0; k < 4; ++k) {
      __builtin_amdgcn_global_store_async_from_lds_b128(
          (gv4i*)(dst + ((lane + 32 * k) << 4)),
          (lv4i*)(buf + ((lane + 32 * k) << 4)), 0, 0);
    }
    WAIT_ASYNC(0);  // buffer reusable
  }
}
#endif  // HAVE_ASYNC_LDS

// ---------------------------------------------------------------------------
// Launch. Inputs per setup_inputs(): d_in[0] = indices (int, 32768 elems),
// d_in[1] = W (float32, 32*1544*512). d_out = [8,4096,512] float32.
// ---------------------------------------------------------------------------
extern "C" void kernel_launch(void* const* d_in, const int* in_sizes, int n_in,
                              void* d_out, int out_size, void* d_ws,
                              size_t ws_size, hipStream_t stream) {
  (void)n_in; (void)d_ws; (void)ws_size; (void)out_size;
  const int*   idx = (const int*)d_in[0];   // integer input -> const int* (per harness contract)
  const float* W   = (const float*)d_in[1];
  float*       out = (float*)d_out;
  const int T = in_sizes[0];  // 8 * 4096 = 32768 tokens

  gather_rows_direct<<<BLOCKS, THREADS, 0, stream>>>(idx, W, out, T);
}